// DAN_ACT_17437567222069
// MI455X (gfx1250) — compile-verified
//
#include <hip/hip_runtime.h>

// Problem constants (reference: V=400000, S=4096, D=300, C=4)
#define S_LEN 4096
#define D_DIM 300
#define C_CLS 4

typedef __attribute__((ext_vector_type(2))) float v2f;
typedef __attribute__((ext_vector_type(8))) float v8f;

// ---------------------------------------------------------------------------
// K1: gather embedding rows into workspace + per-block partial sums for mean
// grid=256 blocks, block=256 threads, each block owns 16 tokens
// ---------------------------------------------------------------------------
__global__ __launch_bounds__(256) void k_gather(const int* __restrict__ idx,
                                                const float* __restrict__ table,
                                                float* __restrict__ embed,
                                                float* __restrict__ pdense) {
  const int tid = threadIdx.x;
  const int b = blockIdx.x;
  const int t0 = b * 16;
  float acc0 = 0.f, acc1 = 0.f;
  for (int t = 0; t < 16; ++t) {
    const int tok = idx[t0 + t];
    const float* __restrict__ src = table + (size_t)tok * D_DIM;
    float* __restrict__ dst = embed + (size_t)(t0 + t) * D_DIM;
    {
      const float val = src[tid];     // tid in [0,255] < 300 always valid
      dst[tid] = val;
      acc0 += val;
    }
    const int d2 = tid + 256;
    if (d2 < D_DIM) {
      const float val = src[d2];
      dst[d2] = val;
      acc1 += val;
    }
  }
  pdense[b * D_DIM + tid] = acc0;
  if (tid + 256 < D_DIM) pdense[b * D_DIM + tid + 256] = acc1;
}

// ---------------------------------------------------------------------------
// K2: reduce 256 partial sums -> dense = mean over S
// ---------------------------------------------------------------------------
__global__ void k_reduce_dense(const float* __restrict__ pdense,
                               float* __restrict__ dense) {
  const int d = threadIdx.x;
  if (d < D_DIM) {
    float s = 0.f;
    for (int b = 0; b < 256; ++b) s += pdense[b * D_DIM + d];
    dense[d] = s * (1.0f / (float)S_LEN);
  }
}

// ---------------------------------------------------------------------------
// K3/K4: y = M(300x300) * x (+ bias).  One row per thread, x staged in LDS.
// ---------------------------------------------------------------------------
__global__ void k_matvec300(const float* __restrict__ M,
                            const float* __restrict__ x,
                            const float* __restrict__ bias,
                            float* __restrict__ y) {
  __shared__ float sx[D_DIM];
  const int tid = threadIdx.x;
  for (int j = tid; j < D_DIM; j += blockDim.x) sx[j] = x[j];
  __syncthreads();
  if (tid < D_DIM) {
    float s = bias ? bias[tid] : 0.f;
    const float* __restrict__ row = M + (size_t)tid * D_DIM;
    for (int j = 0; j < D_DIM; ++j) s = fmaf(row[j], sx[j], s);
    y[tid] = s;
  }
}

// ---------------------------------------------------------------------------
// K5: scores[s] = embed[s,:] . v  via V_WMMA_F32_16X16X4_F32
// One wave per 16-token tile; B = v replicated across all 16 columns so
// every column of D holds the 16 scores. 8 waves/block, 32 blocks = 4096 tok.
// A layout (f32 16x4): lanes 0-15 hold K=0,1 in VGPR0/1; lanes 16-31 K=2,3.
// B layout (f32 4x16): mirror pattern; since all columns equal, lane value is
// v[k + 2*(lane>=16) + vgpr].
// ---------------------------------------------------------------------------
__global__ __launch_bounds__(256) void k_scores(const float* __restrict__ embed,
                                                const float* __restrict__ v,
                                                const float* __restrict__ bb,
                                                float* __restrict__ scores) {
  __shared__ __align__(16) float sv[D_DIM + 4];
  const int tid = threadIdx.x;
  for (int i = tid; i < D_DIM; i += 256) sv[i] = v[i];
  __syncthreads();

  const int wave = tid >> 5;
  const int lane = tid & 31;
  const int base = (blockIdx.x * 8 + wave) * 16;     // token tile base
  const int koff = (lane >> 4) * 2;                  // K half select
  const float* __restrict__ erow = embed + (size_t)(base + (lane & 15)) * D_DIM;

  v8f c = {0.f, 0.f, 0.f, 0.f, 0.f, 0.f, 0.f, 0.f};
  for (int k = 0; k < D_DIM; k += 4) {               // 75 WMMA steps, K=300
    const int ka = k + koff;
    v2f a = *(const v2f*)(erow + ka);                // 8B aligned: 1200B rows
    v2f bm = *(const v2f*)(sv + ka);
    c = __builtin_amdgcn_wmma_f32_16x16x4_f32(false, a, false, bm,
                                              (short)0, c, false, false);
  }

  const float shift = bb[0];
  // D layout: vgpr r, lanes 0-15 -> M=r; lanes 16-31 -> M=r+8 (all N equal)
  if (lane == 0) {
#pragma unroll
    for (int r = 0; r < 8; ++r) scores[base + r] = c[r] + shift;
  } else if (lane == 16) {
#pragma unroll
    for (int r = 0; r < 8; ++r) scores[base + 8 + r] = c[r] + shift;
  }
}

// ---------------------------------------------------------------------------
// K6: softmax over 4096 scores -> reward (written directly to d_out+4)
// ---------------------------------------------------------------------------
__global__ __launch_bounds__(1024) void k_softmax(const float* __restrict__ scores,
                                                  float* __restrict__ reward) {
  __shared__ float buf[1024];
  const int tid = threadIdx.x;
  float m = -3.402823466e38f;
  for (int i = tid; i < S_LEN; i += 1024) m = fmaxf(m, scores[i]);
  buf[tid] = m;
  __syncthreads();
  for (int w = 512; w > 0; w >>= 1) {
    if (tid < w) buf[tid] = fmaxf(buf[tid], buf[tid + w]);
    __syncthreads();
  }
  const float gmax = buf[0];
  __syncthreads();
  float s = 0.f;
  for (int i = tid; i < S_LEN; i += 1024) s += expf(scores[i] - gmax);
  buf[tid] = s;
  __syncthreads();
  for (int w = 512; w > 0; w >>= 1) {
    if (tid < w) buf[tid] += buf[tid + w];
    __syncthreads();
  }
  const float inv = 1.0f / buf[0];
  for (int i = tid; i < S_LEN; i += 1024)
    reward[i] = expf(scores[i] - gmax) * inv;
}

// ---------------------------------------------------------------------------
// K7: partial reward-weighted sums over token chunks (64 tokens per block)
// ---------------------------------------------------------------------------
__global__ void k_wsum(const float* __restrict__ embed,
                       const float* __restrict__ reward,
                       float* __restrict__ padd) {
  const int tid = threadIdx.x;
  const int s0 = blockIdx.x * 64;
  if (tid < D_DIM) {
    float acc = 0.f;
    for (int t = 0; t < 64; ++t)
      acc = fmaf(reward[s0 + t], embed[(size_t)(s0 + t) * D_DIM + tid], acc);
    padd[blockIdx.x * D_DIM + tid] = acc;
  }
}

// ---------------------------------------------------------------------------
// K8: addition = mean of weighted sums; out = (mid + addition) @ W2.T + b2
// ---------------------------------------------------------------------------
__global__ void k_final(const float* __restrict__ padd,
                        const float* __restrict__ mid,
                        const float* __restrict__ W2,
                        const float* __restrict__ b2,
                        float* __restrict__ out) {
  __shared__ float t[D_DIM];
  const int tid = threadIdx.x;
  if (tid < D_DIM) {
    float s = 0.f;
    for (int b = 0; b < 64; ++b) s += padd[b * D_DIM + tid];
    t[tid] = mid[tid] + s * (1.0f / (float)S_LEN);
  }
  __syncthreads();
  if (tid < C_CLS) {
    float s = b2[tid];
    const float* __restrict__ row = W2 + tid * D_DIM;
    for (int j = 0; j < D_DIM; ++j) s = fmaf(row[j], t[j], s);
    out[tid] = s;
  }
}

// ---------------------------------------------------------------------------
extern "C" void kernel_launch(void* const* d_in, const int* in_sizes, int n_in,
                              void* d_out, int out_size, void* d_ws, size_t ws_size,
                              hipStream_t stream) {
  (void)in_sizes; (void)n_in; (void)out_size; (void)ws_size;

  const int*   idx   = (const int*)  d_in[0];  // x_batch  [S]
  const float* table = (const float*)d_in[1];  // emb_table [V,D]
  const float* W1    = (const float*)d_in[2];  // [D,D]
  const float* b1    = (const float*)d_in[3];  // [D]
  const float* Wb    = (const float*)d_in[4];  // [D,D]
  const float* bb    = (const float*)d_in[5];  // [1]
  const float* W2    = (const float*)d_in[6];  // [C,D]
  const float* b2    = (const float*)d_in[7];  // [C]

  float* out    = (float*)d_out;               // [0..3]=logits, [4..4099]=reward
  float* reward = out + C_CLS;

  // workspace carve-up (floats); total ~5.3 MB
  float* ws     = (float*)d_ws;
  float* embed  = ws;                                  // S*D     = 1228800
  float* pdense = embed  + (size_t)S_LEN * D_DIM;      // 256*D   =   76800
  float* dense  = pdense + 256 * D_DIM;                // 512
  float* mid    = dense  + 512;                        // 512
  float* vvec   = mid    + 512;                        // 512
  float* scores = vvec   + 512;                        // S       =    4096
  float* padd   = scores + S_LEN;                      // 64*D    =   19200

  // 1) gather + partial mean pooling
  k_gather<<<256, 256, 0, stream>>>(idx, table, embed, pdense);
  // 2) dense = mean over S
  k_reduce_dense<<<1, 320, 0, stream>>>(pdense, dense);
  // 3) mid = W1 @ dense + b1
  k_matvec300<<<1, 320, 0, stream>>>(W1, dense, b1, mid);
  // 4) v = Wb @ mid
  k_matvec300<<<1, 320, 0, stream>>>(Wb, mid, nullptr, vvec);
  // 5) scores = E @ v + bb   (WMMA f32 16x16x4, one wave per 16 tokens)
  k_scores<<<32, 256, 0, stream>>>(embed, vvec, bb, scores);
  // 6) reward = softmax(scores)  -> written straight into d_out
  k_softmax<<<1, 1024, 0, stream>>>(scores, reward);
  // 7) partial reward-weighted pooling
  k_wsum<<<64, 320, 0, stream>>>(embed, reward, padd);
  // 8) logits = (mid + addition) @ W2.T + b2
  k_final<<<1, 320, 0, stream>>>(padd, mid, W2, b2, out);
}